// QuantumFCL_65481071397670
// MI455X (gfx1250) — compile-verified
//
#include <hip/hip_runtime.h>
#include <hip/hip_bf16.h>

// Quantum FCL, 16 qubits, batch 512 — analytic closed form.
//
// Math: RX gates on a wire fuse (RX(a)RX(b)=RX(a+b)); the pre-CNOT state is a
// product state with P(bit_k = 1) = sin^2(phi_k/2), phi_k = x[b,k]+theta[k].
// The CNOT ring is a basis-state permutation producing prefix XORs:
//   y_w = b_0 ^ ... ^ b_w (w>=1),  y_0 = b_1 ^ ... ^ b_15
// and E[(-1)^(xor of independent bits)] = prod cos(phi_k) over the included set.
// So: out[b,w] = prod_{k<=w} cos(phi_k) (w>=1);  out[b,0] = prod_{k>=1} cos(phi_k).
//
// WMMA mapping: masked product = sign * exp( Llog x Mask^T ), computed with
// chained V_WMMA_F32_16X16X4_F32 (K=16 as 4 steps of K=4). One wave handles a
// 16-sample tile (rows M), columns N = output wire. A second WMMA chain counts
// negative cosines (exact small ints in f32) for the sign parity.
// log|cos| is clamped at -80 so mask zeros never multiply -inf (no NaN), and
// exp(-80) ~ 1.8e-35 ~ 0 matches an exactly-zero cosine.

typedef __attribute__((ext_vector_type(2))) float v2f;
typedef __attribute__((ext_vector_type(8))) float v8f;

#define NQ 16
#define BATCH 512

__global__ __launch_bounds__(256) void qfcl_wmma_kernel(
    const float* __restrict__ x,      // (512,16) row-major
    const float* __restrict__ theta,  // (16)
    float* __restrict__ out)          // (512,16) row-major
{
  const int lane = threadIdx.x & 31;
  const int wave = threadIdx.x >> 5;
  const int tile = blockIdx.x * 8 + wave;   // 16-sample tile, 32 tiles total

  const int m = lane & 15;                  // A-matrix row (sample within tile)
  const int b = tile * 16 + m;              // batch index for this lane's A rows
  const int khalf = (lane >> 4) << 1;       // lanes 0-15 -> K+{0,1}; 16-31 -> K+{2,3}
  const int n = m;                          // B/D column handled by this lane

  // Per-lane A data for the 4 K-chunks: k = c*4 + khalf + j
  float alog[8];
  float aneg[8];
#pragma unroll
  for (int c = 0; c < 4; ++c) {
#pragma unroll
    for (int j = 0; j < 2; ++j) {
      const int k   = c * 4 + khalf + j;
      const float phi = x[b * NQ + k] + theta[k];
      const float cv  = cosf(phi);
      alog[c * 2 + j] = fmaxf(logf(fabsf(cv)), -80.0f);   // log(0)=-inf -> -80
      aneg[c * 2 + j] = (cv < 0.0f) ? 1.0f : 0.0f;
    }
  }

  // Accumulate D = A x B over K in 4 chained WMMAs (C-accumulate), twice:
  // once for log-sums, once for negative-cosine counts.
  v8f acc_log = {};
  v8f acc_cnt = {};
#pragma unroll
  for (int c = 0; c < 4; ++c) {
    const int k0 = c * 4 + khalf;           // B rows this lane supplies
    const int k1 = k0 + 1;
    // Inclusion mask: column n==0 -> k>=1 ; column n>=1 -> k<=n
    const float m0 = (n == 0) ? ((k0 >= 1) ? 1.0f : 0.0f) : ((k0 <= n) ? 1.0f : 0.0f);
    const float m1 = (n == 0) ? ((k1 >= 1) ? 1.0f : 0.0f) : ((k1 <= n) ? 1.0f : 0.0f);
    v2f Bv; Bv.x = m0; Bv.y = m1;

    v2f Al; Al.x = alog[c * 2 + 0]; Al.y = alog[c * 2 + 1];
    v2f An; An.x = aneg[c * 2 + 0]; An.y = aneg[c * 2 + 1];

    acc_log = __builtin_amdgcn_wmma_f32_16x16x4_f32(
        false, Al, false, Bv, (short)0, acc_log, false, false);
    acc_cnt = __builtin_amdgcn_wmma_f32_16x16x4_f32(
        false, An, false, Bv, (short)0, acc_cnt, false, false);
  }

  // D layout (ISA 7.12.2): VGPR r -> lanes 0-15: M=r, lanes 16-31: M=r+8; N = lane%16
  const int mbase = (lane >> 4) << 3;
#pragma unroll
  for (int r = 0; r < 8; ++r) {
    const int row = mbase + r;                      // sample within tile
    const float s   = acc_log[r];
    const float cnt = acc_cnt[r];
    const int parity = ((int)(cnt + 0.5f)) & 1;
    const float mag  = expf(s);
    out[(tile * 16 + row) * NQ + n] = parity ? -mag : mag;
  }
}

extern "C" void kernel_launch(void* const* d_in, const int* in_sizes, int n_in,
                              void* d_out, int out_size, void* d_ws, size_t ws_size,
                              hipStream_t stream) {
  const float* x     = (const float*)d_in[0];   // (512,16) f32
  const float* theta = (const float*)d_in[1];   // (16)     f32
  float* out         = (float*)d_out;           // (512,16) f32

  // 32 tiles of 16 samples; 1 wave32 per tile; 8 waves per block -> 4 blocks.
  qfcl_wmma_kernel<<<4, 256, 0, stream>>>(x, theta, out);
}